// PointLoss_35966056137348
// MI455X (gfx1250) — compile-verified
//
#include <hip/hip_runtime.h>

typedef __attribute__((ext_vector_type(2))) float v2f;
typedef __attribute__((ext_vector_type(8))) float v8f;

#define BATCH 64
#define PH    52
#define PW    720
#define HW    (PH * PW)
#define NPIX  (BATCH * HW)
#define NBLK  512
#define NTHR  256
#define NACC  5
#define RAD2DEG 57.29577951308232f

__global__ __launch_bounds__(NTHR) void ploss_main(
    const float* __restrict__ vmap,   // vertexmap_proj  [B,H,W,3]
    const float* __restrict__ nmapP,  // normalmap_proj  [B,H,W,3]
    const float* __restrict__ unc,    // uncertainty     [B,H,W]
    const float* __restrict__ nowN,   // now_normalmap   [B,H,W,3]
    const float* __restrict__ conf,   // now_confidence  [B,H,W]
    const float* __restrict__ lastN,  // last_normalmap  [B,H,W,3]
    const float* __restrict__ lastV,  // last_vertexmap  [B,H,W,3]
    const int*   __restrict__ nowM,   // now_maskmap     [B,H,W]
    const int*   __restrict__ lastM,  // last_maskmap    [B,H,W]
    float* __restrict__ partials)     // [NBLK][5]
{
    float fov = 0.f, c0 = 0.f, icp = 0.f, ang = 0.f, cn = 0.f;

    for (int i = blockIdx.x * blockDim.x + threadIdx.x; i < NPIX;
         i += gridDim.x * blockDim.x) {
        const int b = i / HW;

        const float vx = vmap[3*i + 0];
        const float vy = vmap[3*i + 1];
        const float vz = vmap[3*i + 2];

        const float nwx = nowN[3*i + 0];
        const float nwy = nowN[3*i + 1];
        const float nwz = nowN[3*i + 2];
        const float nabs = fabsf(nwx) + fabsf(nwy) + fabsf(nwz);
        const bool mask0 = (nowM[i] > 0) && (nabs != 0.0f);

        // spherical re-projection
        const float depth = sqrtf(vx*vx + vy*vy + vz*vz + 1e-8f);
        const float yaw   = atan2f(vy, vx) * RAD2DEG;
        float sz = vz / depth;
        sz = fminf(fmaxf(sz, -1.0f), 1.0f);
        const float pitch = asinf(sz) * RAD2DEG;
        const float px = (180.0f - yaw)   * 2.0f;  // /DH(0.5)
        const float py = (3.0f   - pitch) * 2.0f;  // /DV(0.5)

        // fov loss terms (masked by mask0 only)
        const float m0 = mask0 ? 1.0f : 0.0f;
        const float dx = px - fminf(fmaxf(px, 0.0f), (float)(PW - 1));
        const float dy = py - fminf(fmaxf(py, 0.0f), (float)(PH - 1));
        fov += m0 * (dx*dx + dy*dy);
        c0  += m0;

        // round half-to-even (matches jnp.round)
        const float prx = rintf(px);
        const float pry = rintf(py);
        const bool inb = (prx >= 0.0f) && (prx < (float)PW) &&
                         (pry >= 0.0f) && (pry < (float)PH);

        // gather + residuals only when the pixel can still survive the mask
        if (mask0 && inb && (conf[i] >= 0.5f)) {
            const int prxi = (int)fminf(fmaxf(prx, 0.0f), (float)(PW - 1));
            const int pryi = (int)fminf(fmaxf(pry, 0.0f), (float)(PH - 1));
            const int g = b * HW + pryi * PW + prxi;

            const float nlx = lastN[3*g + 0];
            const float nly = lastN[3*g + 1];
            const float nlz = lastN[3*g + 2];
            const float labs = fabsf(nlx) + fabsf(nly) + fabsf(nlz);

            if ((lastM[g] > 0) && (labs != 0.0f)) {
                const float u = unc[i];

                const float vlx = lastV[3*g + 0];
                const float vly = lastV[3*g + 1];
                const float vlz = lastV[3*g + 2];
                const float r = nlx*(vx - vlx) + nly*(vy - vly) + nlz*(vz - vlz);
                icp += u * fabsf(r);

                const float nnx = nmapP[3*i + 0];
                const float nny = nmapP[3*i + 1];
                const float nnz = nmapP[3*i + 2];
                const float num = fabsf(nlx*nnx + nly*nny + nlz*nnz);
                const float den = sqrtf((nlx*nlx + nly*nly + nlz*nlz) *
                                        (nnx*nnx + nny*nny + nnz*nnz) + 1e-8f);
                ang += u * (1.0f - num / den);
                cn  += 1.0f;
            }
        }
    }

    // deterministic wave32 butterfly reduction (every lane ends with the sum)
    #pragma unroll
    for (int o = 16; o > 0; o >>= 1) {
        fov += __shfl_xor(fov, o, 32);
        c0  += __shfl_xor(c0,  o, 32);
        icp += __shfl_xor(icp, o, 32);
        ang += __shfl_xor(ang, o, 32);
        cn  += __shfl_xor(cn,  o, 32);
    }

    __shared__ float sd[NTHR / 32][NACC];
    const int lane = threadIdx.x & 31;
    const int wv   = threadIdx.x >> 5;
    if (lane == 0) {
        sd[wv][0] = fov; sd[wv][1] = c0; sd[wv][2] = icp;
        sd[wv][3] = ang; sd[wv][4] = cn;
    }
    __syncthreads();
    if (threadIdx.x < NACC) {
        float s = 0.f;
        #pragma unroll
        for (int w = 0; w < NTHR / 32; ++w) s += sd[w][threadIdx.x];
        partials[blockIdx.x * NACC + threadIdx.x] = s;
    }
}

// One-wave finalize: deterministic tree over the 512x5 block partials, then
// the scalar combine total = e^-sx*icp_mean + e^-sq*ang_mean + fov_mean + sx+sq
// done as D = A(16x4) x B(4x16) + C on V_WMMA_F32_16X16X4_F32.
__global__ __launch_bounds__(32) void ploss_final(
    const float* __restrict__ partials,
    const float* __restrict__ sx,
    const float* __restrict__ sq,
    float* __restrict__ out)
{
    const int lane = threadIdx.x;  // exactly one wave32

    float acc[NACC];
    #pragma unroll
    for (int a = 0; a < NACC; ++a) {
        float s = 0.f;
        for (int b = lane; b < NBLK; b += 32) s += partials[b * NACC + a];
        #pragma unroll
        for (int o = 16; o > 0; o >>= 1) s += __shfl_xor(s, o, 32);
        acc[a] = s;  // lane-uniform total
    }

    const float fov_s = acc[0];
    const float n0    = fmaxf(acc[1], 1.0f);
    const float icp_s = acc[2];
    const float ang_s = acc[3];
    const float n     = fmaxf(acc[4], 1.0f);

    const float sx0 = sx[0];
    const float sq0 = sq[0];
    const float icpm = icp_s / n;
    const float angm = ang_s / n;
    const float fovm = fov_s / n0;
    const float wx = expf(-sx0);
    const float wq = expf(-sq0);
    const float bias = sx0 + sq0;

    // A (16x4 f32): lanes 0-15 hold K=0 (v.x) / K=1 (v.y); lanes 16-31 K=2/K=3.
    // B (4x16 f32): v.x lanes0-15 = row K=0, lanes16-31 = row K=1;
    //               v.y lanes0-15 = row K=2, lanes16-31 = row K=3. N = lane%16.
    v2f amat = {0.f, 0.f};
    v2f bmat = {0.f, 0.f};
    v8f cmat = {0.f, 0.f, 0.f, 0.f, 0.f, 0.f, 0.f, 0.f};
    if (lane == 0) {            // A[0][0]=wx, A[0][1]=wq ; B[0][0]=icpm, B[2][0]=fovm
        amat.x = wx; amat.y = wq;
        bmat.x = icpm; bmat.y = fovm;
        cmat[0] = bias;         // C[0][0] = sx + sq
    } else if (lane == 16) {    // A[0][2]=1, A[0][3]=0 ; B[1][0]=angm, B[3][0]=0
        amat.x = 1.0f;
        bmat.x = angm;
    }

    // Full wave executes the WMMA (EXEC all-1s); only the store is divergent.
    v8f d = __builtin_amdgcn_wmma_f32_16x16x4_f32(
        false, amat, false, bmat, (short)0, cmat, false, false);

    if (lane == 0) out[0] = d[0];  // D[0][0] lives in lane 0, VGPR 0
}

extern "C" void kernel_launch(void* const* d_in, const int* in_sizes, int n_in,
                              void* d_out, int out_size, void* d_ws, size_t ws_size,
                              hipStream_t stream) {
    const float* vmap  = (const float*)d_in[0];
    const float* nmapP = (const float*)d_in[1];
    const float* unc   = (const float*)d_in[2];
    const float* nowN  = (const float*)d_in[3];
    const float* conf  = (const float*)d_in[4];
    const float* lastN = (const float*)d_in[5];
    const float* lastV = (const float*)d_in[6];
    const float* sx    = (const float*)d_in[7];
    const float* sq    = (const float*)d_in[8];
    const int*   nowM  = (const int*)d_in[9];
    const int*   lastM = (const int*)d_in[10];

    float* partials = (float*)d_ws;  // NBLK * NACC * 4 = 10 KB

    ploss_main<<<NBLK, NTHR, 0, stream>>>(vmap, nmapP, unc, nowN, conf,
                                          lastN, lastV, nowM, lastM, partials);
    ploss_final<<<1, 32, 0, stream>>>(partials, sx, sq, (float*)d_out);
}